// DoublePropMono_35596688949680
// MI455X (gfx1250) — compile-verified
//
#include <hip/hip_runtime.h>
#include <stdint.h>

#define BATCH 1024
#define W 256
#define H 256
#define DEPTH 8
#define BQ 8                         // batch elements per block (register-resident)
#define HTILE 8                      // h-columns per TDM tile
#define NCHUNK (H / HTILE)           // 32 chunks per depth
#define TOTAL_CHUNKS (DEPTH * NCHUNK)
#define LDS_STRIDE (HTILE + 1)       // TDM pads 1 dword per 8 -> stride 9 (conflict-free)

typedef unsigned int u32;
typedef unsigned long long u64;
typedef __attribute__((ext_vector_type(4))) u32 v4u;
typedef __attribute__((ext_vector_type(8))) int v8i;
typedef __attribute__((ext_vector_type(4))) int v4i;

// Issue one TDM DMA: 256 rows x HTILE contiguous floats (row stride 256 floats in
// global), landing in LDS with +1 dword pad per 8 dwords (hardware padding).
__device__ __forceinline__ void tdm_load_tile(const float* gtile, u32 lds_off) {
  u64 ga = (u64)(uintptr_t)gtile;
  v4u g0;
  g0.x = 1u;                                          // count=1, user descriptor
  g0.y = lds_off;                                     // LDS byte address of tile
  g0.z = (u32)ga;                                     // global_addr[31:0]
  g0.w = (u32)((ga >> 32) & 0x1FFFFFFull) | (2u << 30); // addr[56:32] | type=2
  v8i g1;
  // data_size=2 (4B) | pad_enable | pad_interval=2 (8 dwords) | pad_amount=0 (1 dword)
  g1[0] = (int)((2u << 16) | (1u << 20) | (2u << 22));
  g1[1] = (int)((u32)W << 16);       // tensor_dim0 = 256 (bits 79:48, lo16)
  g1[2] = (int)((u32)W << 16);       // tensor_dim1 = 256 (bits 111:80, lo16)
  g1[3] = (int)((u32)HTILE << 16);   // tile_dim0 = 8 (bits 127:112)
  g1[4] = W;                         // tile_dim1 = 256 rows (bits 143:128)
  g1[5] = W;                         // tensor_dim0_stride = 256 (bits 207:160, lo32)
  g1[6] = 0;
  g1[7] = 0;
  v4i z4 = {0, 0, 0, 0};             // groups 2/3 unused (2-D tensor)
  v8i z8 = {0, 0, 0, 0, 0, 0, 0, 0}; // extra group (6-arg clang-23 form), unused
  __builtin_amdgcn_tensor_load_to_lds(g0, g1, z4, z4, z8, 0);
}

__global__ __launch_bounds__(256)
void doubleprop_kernel(const float* __restrict__ val,
                       const float* __restrict__ M1, const float* __restrict__ B1,
                       const float* __restrict__ M2, const float* __restrict__ B2,
                       float* __restrict__ out) {
  __shared__ float ldsM[2][W * LDS_STRIDE];
  __shared__ float ldsB[2][W * LDS_STRIDE];
  __shared__ float red[8 * BQ];

  const int nb = BATCH / BQ;              // 128 batch-blocks per branch
  const int branch = blockIdx.x / nb;     // 0 -> (M1,B1), 1 -> (M2,B2)
  const int base = (blockIdx.x % nb) * BQ;
  const float* Mp = branch ? M2 : M1;
  const float* Bp = branch ? B2 : B1;

  const int tid = threadIdx.x;            // thread t owns row w = t
  const int lane = tid & 31;
  const int wave = tid >> 5;

  float q[BQ];
#pragma unroll
  for (int i = 0; i < BQ; ++i) q[i] = val[base + i];

  const u32 mOff0 = (u32)(uintptr_t)(void*)&ldsM[0][0];
  const u32 mOff1 = (u32)(uintptr_t)(void*)&ldsM[1][0];
  const u32 bOff0 = (u32)(uintptr_t)(void*)&ldsB[0][0];
  const u32 bOff1 = (u32)(uintptr_t)(void*)&ldsB[1][0];

  // Prefetch chunk 0 into buffer 0. TDM ignores EXEC and issues once per wave,
  // so gate to wave 0 only (uniform branch) to avoid 8x duplicate DMA.
  if (tid < 32) {
    tdm_load_tile(Mp, mOff0);
    tdm_load_tile(Bp, bOff0);
  }

  for (int d = 0; d < DEPTH; ++d) {
    float hmx[BQ];
#pragma unroll
    for (int i = 0; i < BQ; ++i) hmx[i] = -__builtin_inff();

    for (int c = 0; c < NCHUNK; ++c) {
      const int idx = d * NCHUNK + c;
      if (tid < 32) {
        if (idx + 1 < TOTAL_CHUNKS) {
          const int nidx = idx + 1;
          const int nd = nidx >> 5;       // NCHUNK == 32
          const int nc = nidx & 31;
          const float* mg = Mp + (size_t)nd * (W * H) + (size_t)nc * HTILE;
          const float* bg = Bp + (size_t)nd * (W * H) + (size_t)nc * HTILE;
          const int nbuf = nidx & 1;
          tdm_load_tile(mg, nbuf ? mOff1 : mOff0);
          tdm_load_tile(bg, nbuf ? bOff1 : bOff0);
          // 2 outstanding (next chunk) allowed; oldest pair (current chunk) done.
          __builtin_amdgcn_s_wait_tensorcnt(2);
        } else {
          __builtin_amdgcn_s_wait_tensorcnt(0);
        }
      }
      __syncthreads();                    // data for chunk idx visible to all waves

      const int cur = idx & 1;
      const float* mrow = &ldsM[cur][tid * LDS_STRIDE];
      const float* brow = &ldsB[cur][tid * LDS_STRIDE];
#pragma unroll
      for (int h = 0; h < HTILE; ++h) {
        const float mv = mrow[h];
        const float bv = brow[h];
#pragma unroll
        for (int i = 0; i < BQ; ++i)
          hmx[i] = fmaxf(hmx[i], fmaf(q[i], mv, bv));
      }
      __syncthreads();                    // all reads done before buffer reuse
    }

    // hmx[i] is now max over h for row w = tid. min over w: wave min tree
    // (32 rows) then cross-wave min via LDS (8 partials).
#pragma unroll
    for (int i = 0; i < BQ; ++i) {
      float r = hmx[i];
#pragma unroll
      for (int off = 16; off >= 1; off >>= 1)
        r = fminf(r, __shfl_xor(r, off, 32));
      if (lane == 0) red[wave * BQ + i] = r;
    }
    __syncthreads();
#pragma unroll
    for (int i = 0; i < BQ; ++i) {
      float v = red[i];
#pragma unroll
      for (int wv = 1; wv < 8; ++wv) v = fminf(v, red[wv * BQ + i]);
      q[i] = v;                           // q for next depth (uniform)
    }
    __syncthreads();                      // protect `red` before next depth
  }

  if (tid < BQ) out[branch * BATCH + base + tid] = q[tid];
}

extern "C" void kernel_launch(void* const* d_in, const int* in_sizes, int n_in,
                              void* d_out, int out_size, void* d_ws, size_t ws_size,
                              hipStream_t stream) {
  const float* val = (const float*)d_in[0];
  const float* M1  = (const float*)d_in[1];
  const float* B1  = (const float*)d_in[2];
  const float* M2  = (const float*)d_in[3];
  const float* B2  = (const float*)d_in[4];
  float* out = (float*)d_out;

  dim3 grid(2 * (BATCH / BQ));   // 256 workgroups: 2 branches x 128 batch-blocks
  dim3 block(256);               // 8 waves; thread t owns matrix row w = t
  hipLaunchKernelGGL(doubleprop_kernel, grid, block, 0, stream,
                     val, M1, B1, M2, B2, out);
}